// LocalLoss_14336600834797
// MI455X (gfx1250) — compile-verified
//
#include <hip/hip_runtime.h>
#include <math.h>

typedef __attribute__((ext_vector_type(2))) float v2f;
typedef __attribute__((ext_vector_type(8))) float v8f;

#define NB 32
#define ND 768
#define NP 361           // 19*19 pixels
#define NPPAD 368        // padded to 16-multiple (GEMM1 M tiles / GEMM2 K chunks)
#define NL 96
#define MT 23            // ceil(NP/16)
#define NT 6             // NL/16
#define DT 48            // ND/16
#define TEMP_ATT 4.0f
#define TEMP_SIM 5.0f
#define TEMP_LOSS 10.0f
#define EPSV 1e-8f
#define NEGF -3.4028234663852886e38f

__device__ __forceinline__ v8f wmma_f32(v2f a, v2f b, v8f c) {
    // V_WMMA_F32_16X16X4_F32: D = A(16x4) * B(4x16) + C(16x16), fp32 throughout
    return __builtin_amdgcn_wmma_f32_16x16x4_f32(false, a, false, b, (short)0, c,
                                                 false, false);
}

// ---- per-caption word-embedding column norms: wn[i,l] = max(||words[i,:,l]||, eps)
__global__ void wn_kernel(const float* __restrict__ words, float* __restrict__ wn) {
    const int i = blockIdx.x;
    const int l = threadIdx.x;
    if (l >= NL) return;
    const float* w = words + (size_t)i * ND * NL;
    float s = 0.f;
    for (int d = 0; d < ND; ++d) { float v = w[d * NL + l]; s += v * v; }
    wn[i * NL + l] = fmaxf(sqrtf(s), EPSV);
}

// ---- one workgroup per (image j, caption i) pair ----
__global__ void pair_kernel(const float* __restrict__ img,    // (B, D, P)
                            const float* __restrict__ words,  // (B, D, L)
                            const int*   __restrict__ lens,   // (B,)
                            const float* __restrict__ wn,     // (B, L)
                            float* __restrict__ cap,          // (B, B) cap_sim
                            float* __restrict__ out)          // out[2..] = attn maps
{
    const int i    = blockIdx.x;          // caption
    const int j    = blockIdx.y;          // image
    const int tid  = threadIdx.x;
    const int wave = tid >> 5;
    const int lane = tid & 31;
    const int half = lane >> 4;
    const int lm   = lane & 15;
    const int len  = lens[i];

    extern __shared__ float smem[];
    float* S    = smem;                   // NPPAD x NL: scores -> attn_q -> attn_c
    float* numS = smem + NPPAD * NL;      // NL
    float* cn2S = numS + NL;              // NL

    const float* ctx = img   + (size_t)j * ND * NP;   // (D, P)
    const float* wrd = words + (size_t)i * ND * NL;   // (D, L)

    // ========= GEMM1: scores[p,l] = sum_d ctx[d,p]*words[d,l] ====================
    // Register-blocked across full N (6 tiles): one A fragment feeds 6 WMMAs.
    for (int m = wave; m < MT; m += 8) {
        const int p0 = m * 16;
        v8f acc[NT];
#pragma unroll
        for (int nt = 0; nt < NT; ++nt) acc[nt] = (v8f){};
        int pA = p0 + lm; if (pA > NP - 1) pA = NP - 1;   // clamp tail rows
        const float* Abase = ctx + pA;                     // + k*NP
        const float* Bbase = wrd + lm;                     // + k*NL + 16*nt
        for (int k0 = 0; k0 < ND; k0 += 4) {
            const int ka = k0 + 2 * half;
            v2f a;
            a.x = Abase[(size_t)ka * NP];
            a.y = Abase[(size_t)(ka + 1) * NP];
            const float* b0 = Bbase + (size_t)ka * NL;
            const float* b1 = Bbase + (size_t)(ka + 1) * NL;
#pragma unroll
            for (int nt = 0; nt < NT; ++nt) {
                v2f b;
                b.x = b0[16 * nt];
                b.y = b1[16 * nt];
                acc[nt] = wmma_f32(a, b, acc[nt]);
            }
        }
#pragma unroll
        for (int nt = 0; nt < NT; ++nt) {
            const int  lcol    = 16 * nt + lm;
            const bool valid_l = (lcol < len);
#pragma unroll
            for (int r = 0; r < 8; ++r) {
                const int p = p0 + r + 8 * half;
                S[p * NL + lcol] = valid_l ? acc[nt][r] : NEGF;   // token mask
            }
        }
    }
    __syncthreads();

    // ================= row softmax over l -> attn_q ==============================
    for (int p = wave; p < NP; p += 8) {
        float m = NEGF;
        for (int l = lane; l < NL; l += 32) m = fmaxf(m, S[p * NL + l]);
        for (int off = 16; off > 0; off >>= 1) m = fmaxf(m, __shfl_xor(m, off, 32));
        float s = 0.f;
        for (int l = lane; l < NL; l += 32) {
            float e = __expf(S[p * NL + l] - m);
            S[p * NL + l] = e;
            s += e;
        }
        for (int off = 16; off > 0; off >>= 1) s += __shfl_xor(s, off, 32);
        const float inv = 1.f / s;
        for (int l = lane; l < NL; l += 32) S[p * NL + l] *= inv;
    }
    __syncthreads();

    // ============ column softmax over p of attn_q*TEMP_ATT -> attn_c =============
    for (int l = wave; l < NL; l += 8) {
        float m = NEGF;
        for (int p = lane; p < NP; p += 32) m = fmaxf(m, S[p * NL + l] * TEMP_ATT);
        for (int off = 16; off > 0; off >>= 1) m = fmaxf(m, __shfl_xor(m, off, 32));
        float s = 0.f;
        for (int p = lane; p < NP; p += 32) {
            float e = __expf(S[p * NL + l] * TEMP_ATT - m);
            S[p * NL + l] = e;
            s += e;
        }
        for (int off = 16; off > 0; off >>= 1) s += __shfl_xor(s, off, 32);
        const float inv = 1.f / s;
        for (int p = lane; p < NP; p += 32) S[p * NL + l] *= inv;
    }
    __syncthreads();

    // zero K-padding rows (361..367) so GEMM2 chunks of 4 are exact; init reducers
    for (int idx = tid; idx < (NPPAD - NP) * NL; idx += 256) S[NP * NL + idx] = 0.f;
    if (tid < NL) { numS[tid] = 0.f; cn2S[tid] = 0.f; }
    __syncthreads();

    // diagonal attention maps: out[2 + (i*L + l)*P + p] = attn_c[p,l]
    if (i == j) {
        float* maps = out + 2 + (size_t)i * NL * NP;
        for (int idx = tid; idx < NP * NL; idx += 256) {
            const int p = idx / NL, l = idx % NL;
            maps[l * NP + p] = S[idx];
        }
    }

    // ====== GEMM2: wc[d,l] = sum_p ctx[d,p]*attn_c[p,l], fused num/cn2 reduce ====
    // Same N-blocking: one ctx fragment (global) + 6 attn_c fragments (LDS) -> 6 WMMA.
    for (int m = wave; m < DT; m += 8) {
        const int d0 = m * 16;
        v8f acc[NT];
#pragma unroll
        for (int nt = 0; nt < NT; ++nt) acc[nt] = (v8f){};
        const float* Abase = ctx + (size_t)(d0 + lm) * NP;  // row d0+lm of ctx
        const float* Bbase = S + lm;                        // + p*NL + 16*nt (LDS)
        for (int k0 = 0; k0 < NPPAD; k0 += 4) {
            const int ka  = k0 + 2 * half;
            const int ka0 = (ka     < NP) ? ka     : NP - 1;  // clamp (B rows are 0)
            const int ka1 = (ka + 1 < NP) ? ka + 1 : NP - 1;
            v2f a;
            a.x = Abase[ka0];
            a.y = Abase[ka1];
            const float* b0 = Bbase + ka * NL;
            const float* b1 = Bbase + (ka + 1) * NL;
#pragma unroll
            for (int nt = 0; nt < NT; ++nt) {
                v2f b;
                b.x = b0[16 * nt];
                b.y = b1[16 * nt];
                acc[nt] = wmma_f32(a, b, acc[nt]);
            }
        }
#pragma unroll
        for (int nt = 0; nt < NT; ++nt) {
            const int lcol = 16 * nt + lm;
#pragma unroll
            for (int r = 0; r < 8; ++r) {
                const int d  = d0 + r + 8 * half;
                const float wv = acc[nt][r];
                atomicAdd(&numS[lcol], wv * wrd[d * NL + lcol]);  // ds_add_f32
                atomicAdd(&cn2S[lcol], wv * wv);
            }
        }
    }
    __syncthreads();

    // ============ cosine -> sim -> cap_sim[j,i] = log(sum_l sim) =================
    if (tid < NL) {
        float simv = 0.f;
        if (tid < len) {
            const float cn = fmaxf(sqrtf(cn2S[tid]), EPSV);
            const float cs = numS[tid] / (wn[i * NL + tid] * cn);
            simv = __expf(cs * TEMP_SIM);
        }
        numS[tid] = simv;
    }
    __syncthreads();
    if (tid == 0) {
        float s = 0.f;
        for (int l = 0; l < NL; ++l) s += numS[l];
        cap[j * NB + i] = __logf(s);
    }
}

// ---- two 32x32 diagonal cross-entropies, one wave ----
__global__ void loss_kernel(const float* __restrict__ cap, float* __restrict__ out) {
    const int t = threadIdx.x;  // 0..31
    float mx = NEGF;
    for (int i = 0; i < NB; ++i) mx = fmaxf(mx, cap[t * NB + i] * TEMP_LOSS);
    float s = 0.f;
    for (int i = 0; i < NB; ++i) s += __expf(cap[t * NB + i] * TEMP_LOSS - mx);
    float lossr = -(cap[t * NB + t] * TEMP_LOSS - mx - __logf(s));
    float mx2 = NEGF;
    for (int jj = 0; jj < NB; ++jj) mx2 = fmaxf(mx2, cap[jj * NB + t] * TEMP_LOSS);
    float s2 = 0.f;
    for (int jj = 0; jj < NB; ++jj) s2 += __expf(cap[jj * NB + t] * TEMP_LOSS - mx2);
    float lossc = -(cap[t * NB + t] * TEMP_LOSS - mx2 - __logf(s2));
    for (int off = 16; off > 0; off >>= 1) {
        lossr += __shfl_xor(lossr, off, 32);
        lossc += __shfl_xor(lossc, off, 32);
    }
    if (t == 0) { out[0] = lossr / NB; out[1] = lossc / NB; }
}

extern "C" void kernel_launch(void* const* d_in, const int* in_sizes, int n_in,
                              void* d_out, int out_size, void* d_ws, size_t ws_size,
                              hipStream_t stream) {
    (void)in_sizes; (void)n_in; (void)out_size; (void)ws_size;
    const float* img   = (const float*)d_in[0];
    const float* words = (const float*)d_in[1];
    const int*   lens  = (const int*)d_in[2];
    float* out = (float*)d_out;
    float* wn  = (float*)d_ws;            // B*L
    float* cap = wn + NB * NL;            // B*B

    wn_kernel<<<NB, 128, 0, stream>>>(words, wn);

    const size_t shmem = (size_t)(NPPAD * NL + 2 * NL) * sizeof(float);  // ~142 KB
    hipFuncSetAttribute((const void*)pair_kernel,
                        hipFuncAttributeMaxDynamicSharedMemorySize, (int)shmem);
    pair_kernel<<<dim3(NB, NB), 256, shmem, stream>>>(img, words, lens, wn, cap, out);

    loss_kernel<<<1, 32, 0, stream>>>(cap, out);
}